// CubicBSplineGrid4d_68856915689795
// MI455X (gfx1250) — compile-verified
//
#include <hip/hip_runtime.h>
#include <stdint.h>

// Control-point grid resolution (t, d, h, w) and channels.
#define RT 8
#define RD 16
#define RH 16
#define RW 16
#define NC 32
// Padded (linear-extrapolation) resolution.
#define TP (RT + 2)
#define DP (RD + 2)
#define HP (RH + 2)
#define WP (RW + 2)

#define NQ_TOTAL 65536
#define QPB 16                 // queries per (single-wave) block
#define TILE_FLOATS (4 * 4 * 4 * 4 * NC)   // 8192 floats = 32 KB
#define TILE_BYTES (TILE_FLOATS * 4)

typedef unsigned int u32x4 __attribute__((ext_vector_type(4)));
typedef int i32x8 __attribute__((ext_vector_type(8)));
typedef int i32x4 __attribute__((ext_vector_type(4)));

// ---------------------------------------------------------------------------
// Prologue: materialize the padded grid in d_ws.
// Padding per axis: a==0 -> 2*g0 - g1 ; a==N+1 -> 2*g[N-1] - g[N-2]; else g[a-1].
// The pad operator is linear per axis, so the 4-axis composition is the tensor
// product of <=2-tap stencils.
// ---------------------------------------------------------------------------
__device__ __forceinline__ void axis_taps(int a, int n, int* j, float* c) {
  if (a == 0)          { j[0] = 0;     j[1] = 1;     c[0] = 2.0f; c[1] = -1.0f; }
  else if (a == n + 1) { j[0] = n - 1; j[1] = n - 2; c[0] = 2.0f; c[1] = -1.0f; }
  else                 { j[0] = a - 1; j[1] = a - 1; c[0] = 1.0f; c[1] = 0.0f; }
}

__global__ __launch_bounds__(256) void pad_grid_kernel(const float* __restrict__ g,
                                                       float* __restrict__ gp) {
  int idx = blockIdx.x * blockDim.x + threadIdx.x;
  const int total = TP * DP * HP * WP * NC;
  if (idx >= total) return;
  int c = idx & (NC - 1);
  int cell = idx >> 5;
  int wp = cell % WP; cell /= WP;
  int hp = cell % HP; cell /= HP;
  int dp = cell % DP; int tp = cell / DP;

  int jt[2], jd[2], jh[2], jw[2];
  float at[2], ad[2], ah[2], aw[2];
  axis_taps(tp, RT, jt, at);
  axis_taps(dp, RD, jd, ad);
  axis_taps(hp, RH, jh, ah);
  axis_taps(wp, RW, jw, aw);

  float acc = 0.0f;
  for (int a = 0; a < 2; ++a) {
    if (at[a] == 0.0f) continue;
    for (int b = 0; b < 2; ++b) {
      if (ad[b] == 0.0f) continue;
      for (int e = 0; e < 2; ++e) {
        if (ah[e] == 0.0f) continue;
        for (int f = 0; f < 2; ++f) {
          if (aw[f] == 0.0f) continue;
          float cf = at[a] * ad[b] * ah[e] * aw[f];
          int gi = (((jt[a] * RD + jd[b]) * RH + jh[e]) * RW + jw[f]) * NC + c;
          acc += cf * g[gi];
        }
      }
    }
  }
  gp[idx] = acc;
}

// ---------------------------------------------------------------------------
// Main kernel helpers.
// ---------------------------------------------------------------------------
__device__ __forceinline__ void bweights(float t, float* w) {
  float t2 = t * t, t3 = t2 * t;
  const float s = 1.0f / 6.0f;
  w[0] = (1.0f - 3.0f * t + 3.0f * t2 - t3) * s;
  w[1] = (4.0f - 6.0f * t2 + 3.0f * t3) * s;
  w[2] = (1.0f + 3.0f * t + 3.0f * t2 - 3.0f * t3) * s;
  w[3] = t3 * s;
}

__device__ __forceinline__ void query_setup(const float* __restrict__ u, int q,
                                            int& it, int& id, int& ih, int& iw,
                                            float& ft, float& fd, float& fh, float& fw) {
  float4 uu = *(const float4*)(u + q * 4);
  float st = uu.x * (float)(RT - 1);
  float sd = uu.y * (float)(RD - 1);
  float sh = uu.z * (float)(RH - 1);
  float sw = uu.w * (float)(RW - 1);
  float fit = fminf(fmaxf(floorf(st), 0.0f), (float)(RT - 2));
  float fid = fminf(fmaxf(floorf(sd), 0.0f), (float)(RD - 2));
  float fih = fminf(fmaxf(floorf(sh), 0.0f), (float)(RH - 2));
  float fiw = fminf(fmaxf(floorf(sw), 0.0f), (float)(RW - 2));
  ft = st - fit; fd = sd - fid; fh = sh - fih; fw = sw - fiw;
  it = (int)fit; id = (int)fid; ih = (int)fih; iw = (int)fiw;
}

// Issue one TDM tensor_load_to_lds that gathers the 4x4x4x(4*32) f32 tile for
// query q from the padded grid into LDS at byte offset lds_off.
// D# layout per CDNA5 ISA.
__device__ __forceinline__ void issue_tdm(const float* __restrict__ u,
                                          const float* __restrict__ gp,
                                          int q, unsigned lds_off) {
  int it, id, ih, iw; float ft, fd, fh, fw;
  query_setup(u, q, it, id, ih, iw, ft, fd, fh, fw);

  unsigned elem = ((((unsigned)it * DP + (unsigned)id) * HP + (unsigned)ih) * WP
                   + (unsigned)iw) * NC;
  unsigned boff = elem * 4u;
  boff = (unsigned)__builtin_amdgcn_readfirstlane((int)boff);  // lane-uniform
  unsigned long long ga = (unsigned long long)(uintptr_t)gp + (unsigned long long)boff;

  // Group 0: count=1 | lds_addr | global_addr[56:0] | type=2 in [127:126]
  u32x4 g0 = { 1u,
               lds_off,
               (unsigned)ga,
               (((unsigned)(ga >> 32)) & 0x01FFFFFFu) | (2u << 30) };

  // Group 1:
  //  w0: workgroup_mask=0, data_size=2 (4B), no barrier/iterate/pad
  //  tensor_dim0 = WP*NC (merged contiguous w,c axis) @bits[79:48]
  //  tensor_dim1 = HP @bits[111:80]; tile_dim0 = 128 @bits[127:112]
  //  tile_dim1 = 4, tile_dim2 = 4 @bits[159:128]
  //  tensor_dim0_stride = WP*NC @bits[207:160]
  //  tensor_dim1_stride = HP*WP*NC @bits[255:208]
  i32x8 g1 = { (int)(2u << 16),
               (int)(((unsigned)(WP * NC) & 0xFFFFu) << 16),
               (int)((((unsigned)(WP * NC)) >> 16) | (((unsigned)HP & 0xFFFFu) << 16)),
               (int)((((unsigned)HP) >> 16) | (128u << 16)),
               (int)(4u | (4u << 16)),
               (int)(WP * NC),
               (int)((((unsigned)(HP * WP * NC)) & 0xFFFFu) << 16),
               (int)(((unsigned)(HP * WP * NC)) >> 16) };

  // Group 2: tensor_dim2 = DP, tensor_dim3 = TP, tensor_dim2_stride = DP*HP*WP*NC,
  //          tile_dim3 = 4 @bits[127:112]
  i32x4 g2 = { DP, TP, DP * HP * WP * NC, (int)(4u << 16) };

  // Group 3: tensor_dim3_stride (unused dim4 step), tensor_dim4 = 1, tile_dim4 = 0
  i32x4 g3 = { TP * DP * HP * WP * NC, (int)(1u << 16), 0, 0 };

  // Probe-verified 6-arg form (clang-23 / therock headers): the 5th operand is
  // an i32x8 (zero-filled call verified to compile); cpol = 0.
  i32x8 gx = { 0, 0, 0, 0, 0, 0, 0, 0 };
  __builtin_amdgcn_tensor_load_to_lds(g0, g1, g2, g3, gx, 0);
}

// Separable tensor-product reduction from the staged LDS tile.
// lane == channel; each ds_load_b32 reads 32 consecutive floats (bank-free).
__device__ __forceinline__ void compute_query(const float* __restrict__ u, int q,
                                              const float* __restrict__ lbuf, int lane,
                                              float* __restrict__ out) {
  int it, id, ih, iw; float ft, fd, fh, fw;
  query_setup(u, q, it, id, ih, iw, ft, fd, fh, fw);
  float wt[4], wd[4], wh[4], ww[4];
  bweights(ft, wt); bweights(fd, wd); bweights(fh, wh); bweights(fw, ww);

  float acc_t = 0.0f;
#pragma unroll
  for (int a = 0; a < 4; ++a) {
    float acc_d = 0.0f;
#pragma unroll
    for (int b = 0; b < 4; ++b) {
      float acc_h = 0.0f;
#pragma unroll
      for (int e = 0; e < 4; ++e) {
        const float* row = lbuf + (((a * 4 + b) * 4 + e) * 4) * NC + lane;
        float acc_w = ww[0] * row[0 * NC] + ww[1] * row[1 * NC] +
                      ww[2] * row[2 * NC] + ww[3] * row[3 * NC];
        acc_h += wh[e] * acc_w;
      }
      acc_d += wd[b] * acc_h;
    }
    acc_t += wt[a] * acc_d;
  }
  out[q * NC + lane] = acc_t;
}

// ---------------------------------------------------------------------------
// Main kernel: one wave32 per block, lane == channel, QPB queries per block,
// double-buffered TDM (tensorcnt) pipeline. 64 KB dynamic LDS -> 5 waves/WGP.
// ---------------------------------------------------------------------------
__global__ __launch_bounds__(32) void bspline4d_tdm_kernel(const float* __restrict__ u,
                                                           const float* __restrict__ gp,
                                                           float* __restrict__ out) {
  extern __shared__ float smem[];  // 2 * TILE_FLOATS
  const int lane = (int)threadIdx.x;
  const int q0 = (int)blockIdx.x * QPB;
  const unsigned lds_base = (unsigned)(uintptr_t)&smem[0];

  issue_tdm(u, gp, q0, lds_base);  // prologue fill of buffer 0

#pragma unroll 1
  for (int j = 0; j < QPB; ++j) {
    if (j + 1 < QPB) {
      issue_tdm(u, gp, q0 + j + 1,
                lds_base + (unsigned)((j + 1) & 1) * (unsigned)TILE_BYTES);
      __builtin_amdgcn_s_wait_tensorcnt(1);  // older TDM (query j) complete
    } else {
      __builtin_amdgcn_s_wait_tensorcnt(0);
    }
    compute_query(u, q0 + j, smem + (j & 1) * TILE_FLOATS, lane, out);
  }
}

// ---------------------------------------------------------------------------
// Host entry.
// d_in[0] = u (65536 x 4 f32), d_in[1] = grid (8*16*16*16*32 f32)
// d_out   = 65536 x 32 f32
// d_ws    = padded grid scratch (10*18*18*18*32 f32 ~= 7.1 MB)
// ---------------------------------------------------------------------------
extern "C" void kernel_launch(void* const* d_in, const int* in_sizes, int n_in,
                              void* d_out, int out_size, void* d_ws, size_t ws_size,
                              hipStream_t stream) {
  (void)in_sizes; (void)n_in; (void)out_size; (void)ws_size;
  const float* u = (const float*)d_in[0];
  const float* grid = (const float*)d_in[1];
  float* out = (float*)d_out;
  float* gp = (float*)d_ws;

  const int total_pad = TP * DP * HP * WP * NC;
  pad_grid_kernel<<<(total_pad + 255) / 256, 256, 0, stream>>>(grid, gp);

  bspline4d_tdm_kernel<<<NQ_TOTAL / QPB, 32, 2 * TILE_BYTES, stream>>>(u, gp, out);
}